// MyModel_61933428409694
// MI455X (gfx1250) — compile-verified
//
#include <hip/hip_runtime.h>
#include <math.h>

typedef __attribute__((ext_vector_type(2))) float v2f;
typedef __attribute__((ext_vector_type(8))) float v8f;

#define GRAM_WGS     2048
#define WG_THREADS   256
#define WAVES_PER_WG 8

// Stage 1: rank-4 f32 WMMA updates of a 16x16 accumulator whose 10x10 corner is
// the Gram matrix. For x^T x, the A(16x4) and B(4x16) operands are the SAME two
// VGPRs per lane (lane = grp*16 + f ; f = feature, grp = sample-pair of chunk).
// Lanes with f>=10 load a duplicate of feature 0: they pollute only rows/cols
// 10..15 of the accumulator, which are never read -> no masking, no branches.
__global__ void __launch_bounds__(WG_THREADS)
gram_wmma_kernel(const float* __restrict__ x, float* __restrict__ partials,
                 int nchunks, int nwaves)
{
    __shared__ float red[WAVES_PER_WG * 256];
    const int lane   = threadIdx.x & 31;
    const int wgwave = threadIdx.x >> 5;
    const int gwave  = blockIdx.x * WAVES_PER_WG + wgwave;
    const int f      = lane & 15;            // feature/column index (clamped below)
    const int grp    = lane >> 4;            // sample pair {0,1} vs {2,3}
    const int fc     = (f < 10) ? f : 0;     // duplicate feature 0 for pad lanes

    // Per-lane base pointer: chunk c element = xp[c*40] and xp[c*40 + 10]
    const float* xp = x + 20 * grp + fc;

    v8f acc0 = {0.f, 0.f, 0.f, 0.f, 0.f, 0.f, 0.f, 0.f};
    v8f acc1 = {0.f, 0.f, 0.f, 0.f, 0.f, 0.f, 0.f, 0.f};

    const int stride = nwaves;
    int c = gwave;

    // Main loop: 4 chunks (16 rows) per iteration, 8 loads in flight, 4 WMMAs.
    for (; c + 3 * stride < nchunks; c += 4 * stride) {
        const size_t b0 = (size_t)(c             ) * 40;
        const size_t b1 = (size_t)(c +     stride) * 40;
        const size_t b2 = (size_t)(c + 2 * stride) * 40;
        const size_t b3 = (size_t)(c + 3 * stride) * 40;
        v2f t0, t1, t2, t3;
        t0.x = xp[b0];  t0.y = xp[b0 + 10];
        t1.x = xp[b1];  t1.y = xp[b1 + 10];
        t2.x = xp[b2];  t2.y = xp[b2 + 10];
        t3.x = xp[b3];  t3.y = xp[b3 + 10];
        acc0 = __builtin_amdgcn_wmma_f32_16x16x4_f32(false, t0, false, t0, (short)0, acc0, false, false);
        acc1 = __builtin_amdgcn_wmma_f32_16x16x4_f32(false, t1, false, t1, (short)0, acc1, false, false);
        acc0 = __builtin_amdgcn_wmma_f32_16x16x4_f32(false, t2, false, t2, (short)0, acc0, false, false);
        acc1 = __builtin_amdgcn_wmma_f32_16x16x4_f32(false, t3, false, t3, (short)0, acc1, false, false);
    }
    // Tail chunks
    for (; c < nchunks; c += stride) {
        const size_t b0 = (size_t)c * 40;
        v2f t0;
        t0.x = xp[b0];  t0.y = xp[b0 + 10];
        acc0 = __builtin_amdgcn_wmma_f32_16x16x4_f32(false, t0, false, t0, (short)0, acc0, false, false);
    }
    acc0 += acc1;

    // C/D layout: VGPR j holds row M=j (lanes 0-15) / M=j+8 (lanes 16-31), col N=lane&15
    const int Mbase = grp * 8;
#pragma unroll
    for (int j = 0; j < 8; ++j)
        red[wgwave * 256 + (Mbase + j) * 16 + f] = acc0[j];
    __syncthreads();

    // Fixed-order in-workgroup reduction (deterministic), one partial Gram per WG.
    float s = 0.f;
#pragma unroll
    for (int w = 0; w < WAVES_PER_WG; ++w)
        s += red[w * 256 + threadIdx.x];
    partials[(size_t)blockIdx.x * 256 + threadIdx.x] = s;
}

// Stage 2: fixed-order reduction of partials -> 10x10 Gram (+ any tail rows),
// cyclic Jacobi eigendecomposition (ascending eigenvalues, columns like eigh),
// then the tiny MLP + sigmoid transform. O(1) work on lane 0.
__global__ void __launch_bounds__(256)
eig_mlp_kernel(const float* __restrict__ partials, int G,
               const float* __restrict__ x, int tail_start, int nrows,
               const float* __restrict__ W1, const float* __restrict__ b1,
               const float* __restrict__ W2, const float* __restrict__ b2,
               float* __restrict__ out)
{
    __shared__ float gram[256];
    const int t = threadIdx.x;
    float s = 0.f;
    for (int g = 0; g < G; ++g)
        s += partials[(size_t)g * 256 + t];
    gram[t] = s;
    __syncthreads();

    if (t == 0) {
        float A[10][10], V[10][10];
        for (int i = 0; i < 10; ++i)
            for (int j = 0; j < 10; ++j) {
                A[i][j] = gram[i * 16 + j];
                V[i][j] = (i == j) ? 1.f : 0.f;
            }
        // Fold in rows not covered by full 4-row chunks (none when nrows % 4 == 0)
        for (int r = tail_start; r < nrows; ++r)
            for (int i = 0; i < 10; ++i)
                for (int j = 0; j < 10; ++j)
                    A[i][j] += x[(size_t)r * 10 + i] * x[(size_t)r * 10 + j];

        // Cyclic Jacobi sweeps: A <- J^T A J, V <- V J
        for (int sweep = 0; sweep < 20; ++sweep) {
            for (int p = 0; p < 9; ++p) {
                for (int q = p + 1; q < 10; ++q) {
                    float apq = A[p][q];
                    if (fabsf(apq) <= 1e-30f) continue;
                    float theta = (A[q][q] - A[p][p]) / (2.f * apq);
                    float tt = 1.f / (fabsf(theta) + sqrtf(theta * theta + 1.f));
                    if (theta < 0.f) tt = -tt;
                    float cc = 1.f / sqrtf(tt * tt + 1.f);
                    float ss = tt * cc;
                    for (int k = 0; k < 10; ++k) {
                        float akp = A[k][p], akq = A[k][q];
                        A[k][p] = cc * akp - ss * akq;
                        A[k][q] = ss * akp + cc * akq;
                    }
                    for (int k = 0; k < 10; ++k) {
                        float apk = A[p][k], aqk = A[q][k];
                        A[p][k] = cc * apk - ss * aqk;
                        A[q][k] = ss * apk + cc * aqk;
                    }
                    for (int k = 0; k < 10; ++k) {
                        float vkp = V[k][p], vkq = V[k][q];
                        V[k][p] = cc * vkp - ss * vkq;
                        V[k][q] = ss * vkp + cc * vkq;
                    }
                }
            }
        }
        // Sort eigenpairs ascending (eigh convention)
        int   idx[10];
        float ev[10];
        for (int i = 0; i < 10; ++i) { idx[i] = i; ev[i] = A[i][i]; }
        for (int i = 0; i < 9; ++i) {
            int m = i;
            for (int j = i + 1; j < 10; ++j)
                if (ev[j] < ev[m]) m = j;
            float te = ev[i]; ev[i] = ev[m]; ev[m] = te;
            int   ti = idx[i]; idx[i] = idx[m]; idx[m] = ti;
        }
        // y = relu(w @ W1^T + b1) -> [10,16]; y @ W2^T + b2 -> [10,1]; 0.5*(sigmoid+1)
        for (int r = 0; r < 10; ++r) {
            float o = b2[0];
            for (int j = 0; j < 16; ++j) {
                float h = b1[j];
                for (int d = 0; d < 10; ++d)
                    h += V[r][idx[d]] * W1[j * 10 + d];
                h = fmaxf(h, 0.f);
                o += h * W2[j];
            }
            out[r] = 0.5f * (1.f / (1.f + expf(-o)) + 1.f);
        }
    }
}

extern "C" void kernel_launch(void* const* d_in, const int* in_sizes, int n_in,
                              void* d_out, int out_size, void* d_ws, size_t ws_size,
                              hipStream_t stream) {
    const float* x  = (const float*)d_in[0];
    const float* W1 = (const float*)d_in[1];
    const float* b1 = (const float*)d_in[2];
    const float* W2 = (const float*)d_in[3];
    const float* b2 = (const float*)d_in[4];
    float* out      = (float*)d_out;
    float* partials = (float*)d_ws;

    const int D = 10;
    const int nrows   = in_sizes[0] / D;   // 4,000,000
    const int nchunks = nrows / 4;         // full 4-row chunks (WMMA-covered)

    int G = GRAM_WGS;
    size_t need = (size_t)G * 256 * sizeof(float);
    if (need > ws_size) {
        G = (int)(ws_size / (256 * sizeof(float)));
        if (G < 1) G = 1;
    }

    gram_wmma_kernel<<<G, WG_THREADS, 0, stream>>>(x, partials, nchunks,
                                                   G * WAVES_PER_WG);
    eig_mlp_kernel<<<1, 256, 0, stream>>>(partials, G, x, nchunks * 4, nrows,
                                          W1, b1, W2, b2, out);
}